// DeformKernelConv2d_89507118448687
// MI455X (gfx1250) — compile-verified
//
#include <hip/hip_runtime.h>
#include <hip/hip_bf16.h>
#include <stddef.h>

// ---------------------------------------------------------------------------
// Problem constants (from reference): x(4,128,56,56), offset_w(18,128,3,3),
// offset_b(18), weight(128,4,4). K=3, WS=4, DILATION=1.
// ---------------------------------------------------------------------------
#define BATCH 4
#define CH    128
#define HH    56
#define WW    56
#define HWSZ  (HH * WW)        // 3136 (multiple of 32)
#define NPOS  (BATCH * HWSZ)   // 12544
#define NOFF  18               // 2*K*K
#define NTAP  9

typedef __attribute__((ext_vector_type(2))) float v2f;
typedef __attribute__((ext_vector_type(4))) float v4f;
typedef __attribute__((ext_vector_type(8))) float v8f;

// ---------------------------------------------------------------------------
// Kernel 1: pack offset_w (18,128,3,3) into lane-major WMMA B tiles:
//   PB[t][kc/4][lane][j], j = {(K=2h,o=ml),(K=2h+1,o=ml),
//                             (K=2h,o=ml+16),(K=2h+1,o=ml+16)}
// so the conv kernel fetches each lane's B operand pair with ONE b128 load.
// o >= 18 is zero-padded here.
// ---------------------------------------------------------------------------
__global__ void deform_pack_bmat(const float* __restrict__ ow,
                                 float* __restrict__ PB) {
    int i = blockIdx.x * 256 + threadIdx.x;
    if (i >= NTAP * 32 * 32 * 4) return;
    int t    = i / 4096;
    int r    = i - t * 4096;
    int kcg  = r >> 7;            // kc/4: 0..31
    int r2   = r & 127;
    int lane = r2 >> 2;           // 0..31
    int j    = r2 & 3;
    int half = lane >> 4;
    int ml   = lane & 15;
    int c    = kcg * 4 + 2 * half + (j & 1);
    int o    = ml + 16 * (j >> 1);
    PB[i] = (o < NOFF) ? ow[((size_t)o * CH + c) * NTAP + t] : 0.0f;
}

// ---------------------------------------------------------------------------
// Kernel 2: tile-transpose x (B,C,H,W) -> xT[pos][c], pos = (b*H+h)*W+w.
// 32x32 LDS tiles (+1 pad), coalesced reads and writes. HWSZ % 32 == 0 so
// tiles never straddle a batch boundary.
// ---------------------------------------------------------------------------
__global__ __launch_bounds__(256)
void deform_transpose_x(const float* __restrict__ x, float* __restrict__ xT) {
    __shared__ float tile[32][33];
    int bi      = blockIdx.x;          // (NPOS/32) * (CH/32) blocks
    int posTile = bi >> 2;
    int cTile   = bi & 3;
    int pos0    = posTile * 32;
    int c0      = cTile * 32;
    int b  = pos0 / HWSZ;
    int p0 = pos0 - b * HWSZ;
    int lx = threadIdx.x & 31;
    int ly = threadIdx.x >> 5;         // 0..7

    const float* __restrict__ src = x + ((size_t)b * CH + c0) * HWSZ + p0;
    #pragma unroll
    for (int i = 0; i < 4; ++i) {
        int cy = ly + 8 * i;
        tile[cy][lx] = src[(size_t)cy * HWSZ + lx];
    }
    __syncthreads();
    float* __restrict__ dst = xT + (size_t)pos0 * CH + c0;
    #pragma unroll
    for (int i = 0; i < 4; ++i) {
        int row = ly + 8 * i;
        dst[(size_t)row * CH + lx] = tile[lx][row];
    }
}

// ---------------------------------------------------------------------------
// Kernel 3: offset conv as implicit GEMM via V_WMMA_F32_16X16X4_F32.
//   M = B*H*W (16 positions/wave), N = 18 (padded 32), K = 128 ch x 9 taps.
// One wave per block; EXEC all-ones around the WMMAs.
// A (16x4 f32): lane&15 = M; vgpr {x,y} = K{0,1} (lanes 0-15) / K{2,3}
// (lanes 16-31) -> one b64 load from xT. B: one b128 load from PB.
// C/D: vgpr r holds (M = r + 8*(lane>>4), N = lane&15).
// ---------------------------------------------------------------------------
__global__ __launch_bounds__(32)
void deform_offset_conv_wmma(const float* __restrict__ xT,
                             const float* __restrict__ PB,
                             const float* __restrict__ bias,
                             float* __restrict__ offbuf) {
    const int tid  = threadIdx.x;
    const int half = tid >> 4;
    const int ml   = tid & 15;

    int m   = blockIdx.x * 16 + ml;     // NPOS multiple of 16
    int b   = m / HWSZ;
    int rem = m - b * HWSZ;
    int h   = rem / WW;
    int w   = rem - h * WW;

    v8f acc0 = {};    // o = 0..15
    v8f acc1 = {};    // o = 16..31 (16,17 valid)

    #pragma unroll 1
    for (int t = 0; t < NTAP; ++t) {
        const int ky  = t / 3, kx = t - (t / 3) * 3;
        const int yy  = h + ky - 1, xx = w + kx - 1;
        const bool vld = (yy >= 0) & (yy < HH) & (xx >= 0) & (xx < WW);
        const size_t post = ((size_t)b * HH + yy) * WW + xx;
        const float* __restrict__ ap = xT + post * CH + 2 * half;
        const v4f* __restrict__ pb =
            (const v4f*)(PB + (size_t)t * 4096) + tid;

        #pragma unroll 4
        for (int kc = 0; kc < CH; kc += 4) {
            v2f a = {};
            if (vld) a = *(const v2f*)(ap + kc);     // channels ca, ca+1
            v4f bq = pb[(kc >> 2) * 32];             // one b128, coalesced
            v2f b0 = { bq.x, bq.y };                 // o = ml
            v2f b1 = { bq.z, bq.w };                 // o = ml + 16
            acc0 = __builtin_amdgcn_wmma_f32_16x16x4_f32(
                false, a, false, b0, (short)0, acc0, false, false);
            acc1 = __builtin_amdgcn_wmma_f32_16x16x4_f32(
                false, a, false, b1, (short)0, acc1, false, false);
        }
    }

    // Scatter D + bias into offbuf[b][o][h][w].
    const int o0 = ml;
    const int o1 = 16 + ml;
    const float bia0 = bias[o0];
    const float bia1 = (o1 < NOFF) ? bias[o1] : 0.0f;
    #pragma unroll
    for (int r = 0; r < 8; ++r) {
        int mm  = blockIdx.x * 16 + r + 8 * half;
        int bb  = mm / HWSZ;
        int rr  = mm - bb * HWSZ;
        int hh  = rr / WW;
        int ww2 = rr - hh * WW;
        offbuf[(((size_t)bb * NOFF + o0) * HH + hh) * WW + ww2] = acc0[r] + bia0;
        if (o1 < NOFF)
            offbuf[(((size_t)bb * NOFF + o1) * HH + hh) * WW + ww2] =
                acc1[r] + bia1;
    }
}

// ---------------------------------------------------------------------------
// Kernel 4: bilinear sample of the 4x4 per-channel weight table at the
// deformed coordinates, times the 3x3 shifted patch, summed over taps.
// Block = 256 threads = 32 w-positions x 8 channel groups; weight in LDS.
// ---------------------------------------------------------------------------
__global__ __launch_bounds__(256)
void deform_sample_reduce(const float* __restrict__ x,
                          const float* __restrict__ weight,
                          const float* __restrict__ offbuf,
                          float* __restrict__ out) {
    __shared__ float wlds[CH * 16];
    const int tid = threadIdx.x;
    for (int i = tid; i < CH * 16; i += 256) wlds[i] = weight[i];
    __syncthreads();

    const int wi = tid & 31;
    const int cj = tid >> 5;                   // channel group 0..7
    int bi  = blockIdx.x;                      // B * H * 2 blocks
    int b   = bi / (HH * 2);
    int rem = bi - b * (HH * 2);
    int h   = rem >> 1;
    int wt  = rem & 1;
    int w   = wt * 32 + wi;
    const bool active = (w < WW);

    float acc[16];
    #pragma unroll
    for (int i = 0; i < 16; ++i) acc[i] = 0.0f;

    #pragma unroll 1
    for (int k = 0; k < NTAP; ++k) {
        const int ky = k / 3, kx = k - ky * 3;
        size_t obase = (((size_t)b * NOFF + 2 * k) * HH + h) * WW + w;
        float dy = active ? offbuf[obase] : 0.0f;
        float dx = active ? offbuf[obase + HWSZ] : 0.0f;
        // base = arange(3) - 1 + 1.5 -> {0.5, 1.5, 2.5}
        float yc = dy + (0.5f + (float)ky);
        float xc = dx + (0.5f + (float)kx);
        float y0 = floorf(yc);
        float x0 = floorf(xc);

        float cf[4];
        int   id[4];
        #pragma unroll
        for (int cnr = 0; cnr < 4; ++cnr) {
            float yi = y0 + (float)(cnr >> 1);
            float xi = x0 + (float)(cnr & 1);
            float coef = (1.0f - fabsf(yc - yi)) * (1.0f - fabsf(xc - xi));
            bool v = (yi >= 0.0f) & (yi <= 3.0f) & (xi >= 0.0f) & (xi <= 3.0f);
            cf[cnr] = v ? coef : 0.0f;
            int yii = (int)fminf(fmaxf(yi, 0.0f), 3.0f);
            int xii = (int)fminf(fmaxf(xi, 0.0f), 3.0f);
            id[cnr] = yii * 4 + xii;
        }

        const int yy = h + ky - 1, xx = w + kx - 1;
        const bool pv = active & (yy >= 0) & (yy < HH) & (xx >= 0) & (xx < WW);
        const float* __restrict__ xp =
            x + (size_t)b * CH * HWSZ + (size_t)yy * WW + xx;

        #pragma unroll
        for (int ci = 0; ci < 16; ++ci) {
            const int c = cj * 16 + ci;
            const float* wr = wlds + c * 16;
            float wsum = cf[0] * wr[id[0]] + cf[1] * wr[id[1]] +
                         cf[2] * wr[id[2]] + cf[3] * wr[id[3]];
            float xv = pv ? xp[(size_t)c * HWSZ] : 0.0f;
            acc[ci] += wsum * xv;
        }
    }

    if (active) {
        #pragma unroll
        for (int ci = 0; ci < 16; ++ci) {
            const int c = cj * 16 + ci;
            out[(((size_t)b * CH + c) * HH + h) * WW + w] = acc[ci];
        }
    }
}

// ---------------------------------------------------------------------------
// Host-side launcher. Inputs: x, offset_w, offset_b, weight (all f32).
// Workspace layout (floats):
//   [0, 36864)            PB    (9*32*32*4)
//   [36864, 262656)       offbuf(4*18*56*56)
//   [262656, 1868288)     xT    (12544*128)
// Total ~7.5 MB.
// ---------------------------------------------------------------------------
extern "C" void kernel_launch(void* const* d_in, const int* in_sizes, int n_in,
                              void* d_out, int out_size, void* d_ws,
                              size_t ws_size, hipStream_t stream) {
    const float* x        = (const float*)d_in[0];
    const float* offset_w = (const float*)d_in[1];
    const float* offset_b = (const float*)d_in[2];
    const float* weight   = (const float*)d_in[3];
    float*       out      = (float*)d_out;

    float* PB     = (float*)d_ws;                    // 36864 floats
    float* offbuf = PB + NTAP * 32 * 32 * 4;         // 225792 floats
    float* xT     = offbuf + (size_t)BATCH * NOFF * HWSZ;  // 1605632 floats

    // 1) Pack WMMA B tiles (lane-major, zero-padded N).
    {
        int n = NTAP * 32 * 32 * 4;
        deform_pack_bmat<<<(n + 255) / 256, 256, 0, stream>>>(offset_w, PB);
    }
    // 2) Transpose x to position-major for b64 A-operand loads.
    deform_transpose_x<<<(NPOS / 32) * (CH / 32), 256, 0, stream>>>(x, xT);
    // 3) Offset conv (implicit GEMM, f32 WMMA). 12544 rows / 16 = 784 tiles.
    deform_offset_conv_wmma<<<NPOS / 16, 32, 0, stream>>>(
        xT, PB, offset_b, offbuf);
    // 4) Bilinear sampling + patch reduction.
    deform_sample_reduce<<<BATCH * HH * 2, 256, 0, stream>>>(
        x, weight, offbuf, out);
}